// DHGNNLayerDGL_5214090297382
// MI455X (gfx1250) — compile-verified
//
#include <hip/hip_runtime.h>
#include <hip/hip_bf16.h>

// ---------------------------------------------------------------------------
// MI455X / gfx1250 implementation of the DHGNN layer.
//  - fp32 WMMA (V_WMMA_F32_16X16X4_F32) for all projection / gate GEMMs,
//    with both A and B operands staged in LDS (B pair-interleaved so each
//    fragment is a single aligned ds_load_b64 with immediate offset)
//  - L2-resident edge-softmax via GLOBAL_ATOMIC_{MAX_U32, ADD_F32}
//  - wave32 shuffle reductions for LayerNorm
// ---------------------------------------------------------------------------

typedef __attribute__((ext_vector_type(2))) float v2f;
typedef __attribute__((ext_vector_type(8))) float v8f;

#if __has_builtin(__builtin_amdgcn_wmma_f32_16x16x4_f32)
#define HAVE_WMMA_F32X4 1
#else
#define HAVE_WMMA_F32X4 0
#endif

// ---------------- order-preserving float <-> uint for atomic max -----------
__device__ __forceinline__ unsigned f2ord(float f) {
  unsigned u = __float_as_uint(f);
  return (u & 0x80000000u) ? ~u : (u | 0x80000000u);
}
__device__ __forceinline__ float ord2f(unsigned u) {
  return (u & 0x80000000u) ? __uint_as_float(u & 0x7fffffffu)
                           : __uint_as_float(~u);
}
#define ORD_NEG_INF 0x007FFFFFu  // f2ord(-inf)

// ---------------- fill kernels ---------------------------------------------
__global__ void fill_f32_kernel(float* p, float v, int n) {
  int i = blockIdx.x * blockDim.x + threadIdx.x;
  if (i < n) p[i] = v;
}
__global__ void fill_u32_kernel(unsigned* p, unsigned v, int n) {
  int i = blockIdx.x * blockDim.x + threadIdx.x;
  if (i < n) p[i] = v;
}
__global__ void set_has_kernel(const int* __restrict__ idx, int* __restrict__ has, int E) {
  int i = blockIdx.x * blockDim.x + threadIdx.x;
  if (i < E) has[idx[i]] = 1;
}

// ---------------- WMMA GEMM: Y = act( [X0|X1] @ W + bias ) -----------------
// X0/X1: n x 128 row-major (X1 used only when KDIM==256, concatenated on K)
// W: KDIM x 128 row-major.  Block = 256 threads (8 waves).  grid.x = ceil(n/16).
// Each wave owns a 16x16 output tile (cols [16*wave, 16*wave+16)).
// Dynamic LDS: sX = 16 x KDIM (A tile, row-major),
//              sW = (KDIM/2) x 128 x 2 pair-interleaved B tile:
//              sW[(k>>1)*256 + c*2 + (k&1)] = W[k][c]  ->  per-lane B fragment
//              {W[k][c], W[k+1][c]} is one aligned 8B LDS read.
template <int KDIM, bool SIGMOID>
__global__ void wmma_gemm_kernel(const float* __restrict__ X0,
                                 const float* __restrict__ X1,
                                 const float* __restrict__ W,
                                 const float* __restrict__ bias,
                                 float* __restrict__ Y, int n) {
  extern __shared__ float smem[];
  float* sX = smem;               // 16 * KDIM floats
  float* sW = smem + 16 * KDIM;   // KDIM * 128 floats, pair-interleaved

  const int r0 = blockIdx.x * 16;

  // Stage the 16 x KDIM A-tile into LDS (float4, coalesced), zero-pad OOB rows.
  for (int i = threadIdx.x; i < (16 * KDIM) / 4; i += blockDim.x) {
    int elem = i * 4;
    int row = elem / KDIM;
    int kk = elem % KDIM;
    float4 v = make_float4(0.f, 0.f, 0.f, 0.f);
    int gr = r0 + row;
    if (gr < n) {
      const float* sp = (kk < 128) ? (X0 + (size_t)gr * 128 + kk)
                                   : (X1 + (size_t)gr * 128 + (kk - 128));
      v = *(const float4*)sp;
    }
    *(float4*)(sX + row * KDIM + kk) = v;
  }
  // Stage W pair-interleaved (coalesced b128 reads, strided LDS writes).
  for (int i = threadIdx.x; i < (KDIM * 128) / 4; i += blockDim.x) {
    int elem = i * 4;
    int k = elem >> 7;      // row of W
    int c = elem & 127;     // col of W
    float4 v = *(const float4*)(W + (size_t)k * 128 + c);
    float* dp = sW + (k >> 1) * 256 + (k & 1) + c * 2;
    dp[0] = v.x; dp[2] = v.y; dp[4] = v.z; dp[6] = v.w;
  }
  __syncthreads();

  const int lane = threadIdx.x & 31;
  const int wave = threadIdx.x >> 5;
  const int c0 = wave * 16;          // output column tile
  const int m = lane & 15;           // A-row / B-col index for this lane
  const int khalf = (lane < 16) ? 0 : 2;

  const float* aP = sX + m * KDIM + khalf;          // A: +k0 per step
  const float* bP = sW + (khalf >> 1) * 256 + (c0 + m) * 2;  // B: +(k0>>1)*256

  v8f acc = {0.f, 0.f, 0.f, 0.f, 0.f, 0.f, 0.f, 0.f};

#pragma unroll 4
  for (int k0 = 0; k0 < KDIM; k0 += 4) {
    // A fragment: 16x4 fp32, rows in lanes, K split across lane halves
    v2f a = *(const v2f*)(aP + k0);
    // B fragment: 4x16 fp32, cols in lanes, K split across lane halves
    v2f b = *(const v2f*)(bP + (k0 >> 1) * 256);
#if HAVE_WMMA_F32X4
    acc = __builtin_amdgcn_wmma_f32_16x16x4_f32(
        /*neg_a=*/false, a, /*neg_b=*/false, b,
        /*c_mod=*/(short)0, acc, /*reuse_a=*/false, /*reuse_b=*/false);
#else
    // compile-safety fallback (matrix pipe unavailable): rank-2 FMA sketch
#pragma unroll
    for (int i = 0; i < 8; ++i) acc[i] += a[0] * b[0] + a[1] * b[1];
#endif
  }

  // Epilogue: lane holds column c0+m for rows rbase..rbase+7
  const float bv = bias[c0 + m];
  const int rbase = r0 + ((lane < 16) ? 0 : 8);
#pragma unroll
  for (int i = 0; i < 8; ++i) {
    int gr = rbase + i;
    if (gr < n) {
      float y = acc[i] + bv;
      if (SIGMOID) y = 1.f / (1.f + expf(-y));
      Y[(size_t)gr * 128 + c0 + m] = y;
    }
  }
}

// ---------------- edge-softmax kernels -------------------------------------
// One thread per (edge, head): 32-element dot over a 128B head chunk.
__global__ void attn_score_kernel(const float* __restrict__ Km,
                                  const float* __restrict__ Qm,
                                  const int* __restrict__ src,
                                  const int* __restrict__ dst,
                                  float* __restrict__ score,
                                  unsigned* __restrict__ smax, int E) {
  int idx = blockIdx.x * blockDim.x + threadIdx.x;
  if (idx >= E * 4) return;
  int e = idx >> 2, h = idx & 3;
  int s = src[e], d = dst[e];
  const float4* kp = (const float4*)(Km + (size_t)s * 128 + h * 32);
  const float4* qp = (const float4*)(Qm + (size_t)d * 128 + h * 32);
  float acc = 0.f;
#pragma unroll
  for (int i = 0; i < 8; ++i) {
    float4 kv = kp[i], qv = qp[i];
    acc += kv.x * qv.x + kv.y * qv.y + kv.z * qv.z + kv.w * qv.w;
  }
  acc *= 0.17677669529663687f;  // 1/sqrt(32)
  score[idx] = acc;
  atomicMax(&smax[d * 4 + h], f2ord(acc));
}

__global__ void attn_exp_kernel(const int* __restrict__ dst,
                                const float* __restrict__ score,
                                const unsigned* __restrict__ smax,
                                float* __restrict__ aw,
                                float* __restrict__ denom, int E) {
  int idx = blockIdx.x * blockDim.x + threadIdx.x;
  if (idx >= E * 4) return;
  int e = idx >> 2, h = idx & 3;
  int d = dst[e];
  float a = expf(score[idx] - ord2f(smax[d * 4 + h]));
  aw[idx] = a;
  atomicAdd(&denom[d * 4 + h], a);
}

// One thread per (edge, feature-dim): coalesced V-row read, f32 atomic scatter.
__global__ void attn_scatter_kernel(const float* __restrict__ Vm,
                                    const int* __restrict__ src,
                                    const int* __restrict__ dst,
                                    const float* __restrict__ aw,
                                    const float* __restrict__ denom,
                                    float* __restrict__ out, int E) {
  int idx = blockIdx.x * blockDim.x + threadIdx.x;
  if (idx >= E * 128) return;
  int e = idx >> 7, c = idx & 127, h = c >> 5;
  int s = src[e], d = dst[e];
  float w = aw[e * 4 + h] / denom[d * 4 + h];
  atomicAdd(&out[(size_t)d * 128 + c], Vm[(size_t)s * 128 + c] * w);
}

// ---------------- gating ----------------------------------------------------
__global__ void gated_combine_kernel(const float* __restrict__ za,
                                     const float* __restrict__ zb,
                                     const float* __restrict__ ma,
                                     const float* __restrict__ mb,
                                     const int* __restrict__ hasA,
                                     const int* __restrict__ hasB,
                                     float* __restrict__ out, int n) {
  int idx = blockIdx.x * blockDim.x + threadIdx.x;
  if (idx >= n * 128) return;
  int i = idx >> 7;
  float a = za[idx], b = zb[idx];
  float va = ma[idx], vb = mb[idx];
  float zs = a + b + 1e-8f;
  float both = a / zs * va + b / zs * vb;
  bool ha = hasA[i] != 0, hb = hasB[i] != 0;
  out[idx] = (ha && hb) ? both : (ha ? va : (hb ? vb : 0.f));
}

// ---------------- LayerNorm (wave32 per row) -------------------------------
__global__ void ln_kernel(const float* __restrict__ x,
                          const float* __restrict__ res,  // optional residual
                          const float* __restrict__ g,
                          const float* __restrict__ b,
                          float* __restrict__ y, int n) {
  int wid = threadIdx.x >> 5, lane = threadIdx.x & 31;
  int row = blockIdx.x * (blockDim.x >> 5) + wid;
  if (row >= n) return;
  float4 v = ((const float4*)(x + (size_t)row * 128))[lane];
  if (res) {
    float4 r = ((const float4*)(res + (size_t)row * 128))[lane];
    v.x += r.x; v.y += r.y; v.z += r.z; v.w += r.w;
  }
  float s = v.x + v.y + v.z + v.w;
#pragma unroll
  for (int o = 16; o > 0; o >>= 1) s += __shfl_xor(s, o, 32);
  float mu = s * 0.0078125f;  // /128
  float dx = v.x - mu, dy = v.y - mu, dz = v.z - mu, dw = v.w - mu;
  float q = dx * dx + dy * dy + dz * dz + dw * dw;
#pragma unroll
  for (int o = 16; o > 0; o >>= 1) q += __shfl_xor(q, o, 32);
  float rs = rsqrtf(q * 0.0078125f + 1e-5f);
  float4 gv = ((const float4*)g)[lane];
  float4 bv = ((const float4*)b)[lane];
  float4 o4;
  o4.x = dx * rs * gv.x + bv.x;
  o4.y = dy * rs * gv.y + bv.y;
  o4.z = dz * rs * gv.z + bv.z;
  o4.w = dw * rs * gv.w + bv.w;
  ((float4*)(y + (size_t)row * 128))[lane] = o4;
}

// ---------------------------------------------------------------------------
// host-side orchestration
// ---------------------------------------------------------------------------
namespace {

struct AttnP {
  const float *wq, *bq, *wk, *bk, *wv, *bv;
};

inline dim3 g1(long total, int blk) { return dim3((unsigned)((total + blk - 1) / blk)); }

constexpr size_t kSmem128 = (size_t)(16 * 128 + 128 * 128) * sizeof(float);  // 72 KB
constexpr size_t kSmem256 = (size_t)(16 * 256 + 256 * 128) * sizeof(float);  // 144 KB

void run_attn(const float* hsrc, int nsrc, const float* hdst, int ndst,
              const AttnP& p, const int* src, const int* dst, int E,
              float* Q, float* Kb, float* Vb, float* score, unsigned* smax,
              float* denom, float* outbuf, hipStream_t st) {
  fill_u32_kernel<<<g1(ndst * 4, 256), 256, 0, st>>>(smax, ORD_NEG_INF, ndst * 4);
  fill_f32_kernel<<<g1(ndst * 4, 256), 256, 0, st>>>(denom, 0.f, ndst * 4);
  fill_f32_kernel<<<g1((long)ndst * 128, 256), 256, 0, st>>>(outbuf, 0.f, ndst * 128);

  wmma_gemm_kernel<128, false><<<dim3((ndst + 15) / 16), 256, kSmem128, st>>>(hdst, hdst, p.wq, p.bq, Q, ndst);
  wmma_gemm_kernel<128, false><<<dim3((nsrc + 15) / 16), 256, kSmem128, st>>>(hsrc, hsrc, p.wk, p.bk, Kb, nsrc);
  wmma_gemm_kernel<128, false><<<dim3((nsrc + 15) / 16), 256, kSmem128, st>>>(hsrc, hsrc, p.wv, p.bv, Vb, nsrc);

  attn_score_kernel<<<g1((long)E * 4, 256), 256, 0, st>>>(Kb, Q, src, dst, score, smax, E);
  attn_exp_kernel<<<g1((long)E * 4, 256), 256, 0, st>>>(dst, score, smax, score, denom, E);
  attn_scatter_kernel<<<g1((long)E * 128, 256), 256, 0, st>>>(Vb, src, dst, score, denom, outbuf, E);
}

}  // namespace

extern "C" void kernel_launch(void* const* d_in, const int* in_sizes, int n_in,
                              void* d_out, int out_size, void* d_ws, size_t ws_size,
                              hipStream_t stream) {
  (void)n_in; (void)out_size; (void)ws_size;

  const float* h_drug = (const float*)d_in[0];
  const float* h_dis  = (const float*)d_in[1];
  const float* h_gene = (const float*)d_in[2];
  const float* h_ph   = (const float*)d_in[3];
  const int* src_dg = (const int*)d_in[4];
  const int* dst_dg = (const int*)d_in[5];
  const int* src_sg = (const int*)d_in[6];
  const int* dst_sg = (const int*)d_in[7];
  const int* src_dp = (const int*)d_in[8];
  const int* dst_dp = (const int*)d_in[9];
  const int* src_sp = (const int*)d_in[10];
  const int* dst_sp = (const int*)d_in[11];

  const int nDrug  = in_sizes[0] / 128;
  const int nDis   = in_sizes[1] / 128;
  const int nGene  = in_sizes[2] / 128;
  const int nPheno = in_sizes[3] / 128;
  const int E = in_sizes[4];
  const int nMax = nGene > nPheno ? nGene : nPheno;  // largest node table

  auto P = [&](int i) { return (const float*)d_in[i]; };
  auto AT = [&](int base) {  // _attn_params flattened as q.w,q.b,k.w,k.b,v.w,v.b
    AttnP a{P(base), P(base + 1), P(base + 2), P(base + 3), P(base + 4), P(base + 5)};
    return a;
  };
  AttnP n2e_dg = AT(12), n2e_sg = AT(18), n2e_dp = AT(24), n2e_sp = AT(30);
  AttnP e2n_gd = AT(36), e2n_pd = AT(42), e2n_gs = AT(48), e2n_ps = AT(54);
  const float *w_gdrug = P(60), *b_gdrug = P(61);
  const float *w_gdis  = P(62), *b_gdis  = P(63);
  const float *w_ggene = P(64), *b_ggene = P(65);
  const float *w_gph   = P(66), *b_gph   = P(67);
  const float *ln1g = P(68), *ln1b = P(69), *ln2g = P(70), *ln2b = P(71);

  // ---- workspace carve-up (fp32 elements) ----
  float* ws = (float*)d_ws;
  size_t off = 0;
  auto alloc = [&](size_t nelem) { float* p = ws + off; off += nelem; return p; };
  const size_t ROW = (size_t)nMax * 128;
  float* Q     = alloc(ROW);
  float* Kb    = alloc(ROW);
  float* Vb    = alloc(ROW);
  float* mA    = alloc(ROW);
  float* mB    = alloc(ROW);
  float* za    = alloc(ROW);
  float* zb    = alloc(ROW);
  float* gout  = alloc(ROW);
  float* score = alloc((size_t)E * 4);
  float* denom = alloc((size_t)nMax * 4);
  unsigned* smax = (unsigned*)alloc((size_t)nMax * 4);
  int* hasA = (int*)alloc(nMax);
  int* hasB = (int*)alloc(nMax);

  float* out = (float*)d_out;
  float* out_drug  = out;
  float* out_dis   = out + (size_t)nDrug * 128;
  float* out_gene  = out + (size_t)(nDrug + nDis) * 128;
  float* out_pheno = out + (size_t)(nDrug + nDis + nGene) * 128;
  hipStream_t st = stream;

  // ================= Round 1: node -> gene/pheno =================
  // gene
  run_attn(h_drug, nDrug, h_gene, nGene, n2e_dg, src_dg, dst_dg, E, Q, Kb, Vb, score, smax, denom, mA, st);
  run_attn(h_dis,  nDis,  h_gene, nGene, n2e_sg, src_sg, dst_sg, E, Q, Kb, Vb, score, smax, denom, mB, st);
  fill_u32_kernel<<<g1(nGene, 256), 256, 0, st>>>((unsigned*)hasA, 0u, nGene);
  fill_u32_kernel<<<g1(nGene, 256), 256, 0, st>>>((unsigned*)hasB, 0u, nGene);
  set_has_kernel<<<g1(E, 256), 256, 0, st>>>(dst_dg, hasA, E);
  set_has_kernel<<<g1(E, 256), 256, 0, st>>>(dst_sg, hasB, E);
  wmma_gemm_kernel<256, true><<<dim3((nGene + 15) / 16), 256, kSmem256, st>>>(mA, mB, w_gdrug, b_gdrug, za, nGene);
  wmma_gemm_kernel<256, true><<<dim3((nGene + 15) / 16), 256, kSmem256, st>>>(mA, mB, w_gdis, b_gdis, zb, nGene);
  gated_combine_kernel<<<g1((long)nGene * 128, 256), 256, 0, st>>>(za, zb, mA, mB, hasA, hasB, out_gene, nGene);
  ln_kernel<<<g1(nGene, 8) /*8 rows/block*/, 256, 0, st>>>(out_gene, nullptr, ln1g, ln1b, out_gene, nGene);

  // pheno
  run_attn(h_drug, nDrug, h_ph, nPheno, n2e_dp, src_dp, dst_dp, E, Q, Kb, Vb, score, smax, denom, mA, st);
  run_attn(h_dis,  nDis,  h_ph, nPheno, n2e_sp, src_sp, dst_sp, E, Q, Kb, Vb, score, smax, denom, mB, st);
  fill_u32_kernel<<<g1(nPheno, 256), 256, 0, st>>>((unsigned*)hasA, 0u, nPheno);
  fill_u32_kernel<<<g1(nPheno, 256), 256, 0, st>>>((unsigned*)hasB, 0u, nPheno);
  set_has_kernel<<<g1(E, 256), 256, 0, st>>>(dst_dp, hasA, E);
  set_has_kernel<<<g1(E, 256), 256, 0, st>>>(dst_sp, hasB, E);
  wmma_gemm_kernel<256, true><<<dim3((nPheno + 15) / 16), 256, kSmem256, st>>>(mA, mB, w_gdrug, b_gdrug, za, nPheno);
  wmma_gemm_kernel<256, true><<<dim3((nPheno + 15) / 16), 256, kSmem256, st>>>(mA, mB, w_gdis, b_gdis, zb, nPheno);
  gated_combine_kernel<<<g1((long)nPheno * 128, 256), 256, 0, st>>>(za, zb, mA, mB, hasA, hasB, out_pheno, nPheno);
  ln_kernel<<<g1(nPheno, 8), 256, 0, st>>>(out_pheno, nullptr, ln1g, ln1b, out_pheno, nPheno);

  // ================= Round 2: gene/pheno -> drug/dis =================
  // drug (edges reversed: src index array = dst_*, dst index array = src_*)
  run_attn(out_gene,  nGene,  h_drug, nDrug, e2n_gd, dst_dg, src_dg, E, Q, Kb, Vb, score, smax, denom, mA, st);
  run_attn(out_pheno, nPheno, h_drug, nDrug, e2n_pd, dst_dp, src_dp, E, Q, Kb, Vb, score, smax, denom, mB, st);
  fill_u32_kernel<<<g1(nDrug, 256), 256, 0, st>>>((unsigned*)hasA, 0u, nDrug);
  fill_u32_kernel<<<g1(nDrug, 256), 256, 0, st>>>((unsigned*)hasB, 0u, nDrug);
  set_has_kernel<<<g1(E, 256), 256, 0, st>>>(src_dg, hasA, E);
  set_has_kernel<<<g1(E, 256), 256, 0, st>>>(src_dp, hasB, E);
  wmma_gemm_kernel<256, true><<<dim3((nDrug + 15) / 16), 256, kSmem256, st>>>(mA, mB, w_ggene, b_ggene, za, nDrug);
  wmma_gemm_kernel<256, true><<<dim3((nDrug + 15) / 16), 256, kSmem256, st>>>(mA, mB, w_gph, b_gph, zb, nDrug);
  gated_combine_kernel<<<g1((long)nDrug * 128, 256), 256, 0, st>>>(za, zb, mA, mB, hasA, hasB, gout, nDrug);
  ln_kernel<<<g1(nDrug, 8), 256, 0, st>>>(gout, h_drug, ln2g, ln2b, out_drug, nDrug);

  // disease
  run_attn(out_gene,  nGene,  h_dis, nDis, e2n_gs, dst_sg, src_sg, E, Q, Kb, Vb, score, smax, denom, mA, st);
  run_attn(out_pheno, nPheno, h_dis, nDis, e2n_ps, dst_sp, src_sp, E, Q, Kb, Vb, score, smax, denom, mB, st);
  fill_u32_kernel<<<g1(nDis, 256), 256, 0, st>>>((unsigned*)hasA, 0u, nDis);
  fill_u32_kernel<<<g1(nDis, 256), 256, 0, st>>>((unsigned*)hasB, 0u, nDis);
  set_has_kernel<<<g1(E, 256), 256, 0, st>>>(src_sg, hasA, E);
  set_has_kernel<<<g1(E, 256), 256, 0, st>>>(src_sp, hasB, E);
  wmma_gemm_kernel<256, true><<<dim3((nDis + 15) / 16), 256, kSmem256, st>>>(mA, mB, w_ggene, b_ggene, za, nDis);
  wmma_gemm_kernel<256, true><<<dim3((nDis + 15) / 16), 256, kSmem256, st>>>(mA, mB, w_gph, b_gph, zb, nDis);
  gated_combine_kernel<<<g1((long)nDis * 128, 256), 256, 0, st>>>(za, zb, mA, mB, hasA, hasB, gout, nDis);
  ln_kernel<<<g1(nDis, 8), 256, 0, st>>>(gout, h_dis, ln2g, ln2b, out_dis, nDis);
}